// TokenSubNet_79998060855895
// MI455X (gfx1250) — compile-verified
//
#include <hip/hip_runtime.h>
#include <math.h>

typedef __attribute__((ext_vector_type(2))) float v2f;
typedef __attribute__((ext_vector_type(8))) float v8f;

#define BATCH  2
#define LTOK   4096
#define EMBED  192
#define DIN    384
#define DSTATE 16
#define DTRANK 12
#define DCONV  4
#define XZLD   768   // 2*DIN, xz row stride
#define DBCLD  48    // padded dbc row stride (valid cols: 0..43)

__device__ __forceinline__ v8f wmma4(v2f a, v2f b, v8f c) {
    return __builtin_amdgcn_wmma_f32_16x16x4_f32(false, a, false, b,
                                                 (short)0, c, false, false);
}

// ---------------------------------------------------------------------------
// Fast fp32 WMMA GEMM (no guards):  C[M x N] = A @ W (+bias) (+residual)
// Requires K % 8 == 0, N % 32 == 0, M % 32 == 0.
// One wave computes a 32x32 C tile: 2 M-tiles x 2 N-tiles = 4 independent
// accumulator chains. Per K-8 iteration: 4 b64 A-loads + 8 b32 W-loads feed
// 8 WMMAs (A frags shared across N-tiles, W frags shared across M-tiles).
// ---------------------------------------------------------------------------
__global__ void wmma_gemm_f32_4t(const float* __restrict__ A, int lda,
                                 const float* __restrict__ W, int ldw,
                                 const float* __restrict__ bias,
                                 const float* __restrict__ residual, int ldr,
                                 float* __restrict__ C, int ldc,
                                 int M, int N, int K)
{
    const int wave  = (blockIdx.x * blockDim.x + threadIdx.x) >> 5;
    const int lane  = threadIdx.x & 31;
    const int ntn   = N >> 5;                   // 32-col tiles
    const int total = (M >> 5) * ntn;
    if (wave >= total) return;                  // uniform per wave
    const int tm = wave / ntn;
    const int tn = wave % ntn;

    const int lg   = lane >> 4;                 // half-wave (K sub-pair)
    const int ll   = lane & 15;
    const int row0 = (tm << 5) + ll;            // A row, first M-tile
    const int row1 = row0 + 16;                 // A row, second M-tile
    const int col0 = (tn << 5) + ll;            // C col, first N-tile
    const int col1 = col0 + 16;                 // C col, second N-tile

    v8f c00 = {}, c01 = {}, c10 = {}, c11 = {};
    if (bias) {
        const float b0 = bias[col0], b1 = bias[col1];
        #pragma unroll
        for (int v = 0; v < 8; ++v) {
            c00[v] = b0; c01[v] = b1; c10[v] = b0; c11[v] = b1;
        }
    }

    const float* arow0 = A + (size_t)row0 * lda;
    const float* arow1 = A + (size_t)row1 * lda;
    const float* wc0   = W + col0;
    const float* wc1   = W + col1;

    for (int k = 0; k < K; k += 8) {
        if (k + 64 < K) {
            __builtin_prefetch(arow0 + k + 64, 0, 3);
            __builtin_prefetch(arow1 + k + 64, 0, 3);
        }
        const int ka = k + 2 * lg;              // this lane's K offset
        const v2f a0m0 = *(const v2f*)(arow0 + ka);      // K = ka, ka+1
        const v2f a0m1 = *(const v2f*)(arow1 + ka);
        const v2f a1m0 = *(const v2f*)(arow0 + ka + 4);  // K = ka+4, ka+5
        const v2f a1m1 = *(const v2f*)(arow1 + ka + 4);
        v2f b00, b10, b01, b11;
        b00.x = wc0[(size_t)ka * ldw];       b00.y = wc0[(size_t)(ka + 1) * ldw];
        b10.x = wc1[(size_t)ka * ldw];       b10.y = wc1[(size_t)(ka + 1) * ldw];
        b01.x = wc0[(size_t)(ka + 4) * ldw]; b01.y = wc0[(size_t)(ka + 5) * ldw];
        b11.x = wc1[(size_t)(ka + 4) * ldw]; b11.y = wc1[(size_t)(ka + 5) * ldw];
        c00 = wmma4(a0m0, b00, c00);
        c01 = wmma4(a0m0, b10, c01);
        c10 = wmma4(a0m1, b00, c10);
        c11 = wmma4(a0m1, b10, c11);
        c00 = wmma4(a1m0, b01, c00);
        c01 = wmma4(a1m0, b11, c01);
        c10 = wmma4(a1m1, b01, c10);
        c11 = wmma4(a1m1, b11, c11);
    }

    // C/D layout: VGPR v -> row (tile_m*16) + v + 8*lg
    const int crow0 = (tm << 5) + (lg << 3);        // first M-tile
    const int crow1 = crow0 + 16;                   // second M-tile
    #pragma unroll
    for (int v = 0; v < 8; ++v) {
        float o00 = c00[v], o01 = c01[v], o10 = c10[v], o11 = c11[v];
        if (residual) {
            o00 += residual[(size_t)(crow0 + v) * ldr + col0];
            o01 += residual[(size_t)(crow0 + v) * ldr + col1];
            o10 += residual[(size_t)(crow1 + v) * ldr + col0];
            o11 += residual[(size_t)(crow1 + v) * ldr + col1];
        }
        C[(size_t)(crow0 + v) * ldc + col0] = o00;
        C[(size_t)(crow0 + v) * ldc + col1] = o01;
        C[(size_t)(crow1 + v) * ldc + col0] = o10;
        C[(size_t)(crow1 + v) * ldc + col1] = o11;
    }
}

// ---------------------------------------------------------------------------
// Guarded fp32 WMMA GEMM for the padded small GEMMs (x_proj 384->44-of-48,
// dt 12-of-16 -> 384). Branchless: indices clamped, contributions masked, so
// loads stay unconditional (no exec-mask juggling in the loop).
//   KA : valid K (A cols / W rows >= KA contribute 0)
//   NW : valid W cols (cols >= NW contribute 0; C buffer padded to N)
//   act: 0 = none, 1 = softplus
// ---------------------------------------------------------------------------
__global__ void wmma_gemm_f32_guard(const float* __restrict__ A, int lda,
                                    const float* __restrict__ W, int ldw,
                                    const float* __restrict__ bias,
                                    float* __restrict__ C, int ldc,
                                    int M, int N, int Kpad, int KA, int NW,
                                    int act)
{
    const int wave  = (blockIdx.x * blockDim.x + threadIdx.x) >> 5;
    const int lane  = threadIdx.x & 31;
    const int ntn   = N >> 4;
    const int total = (M >> 4) * ntn;
    if (wave >= total) return;
    const int tm = wave / ntn;
    const int tn = wave % ntn;

    const int lg  = lane >> 4;
    const int ll  = lane & 15;
    const int row = (tm << 4) + ll;
    const int col = (tn << 4) + ll;

    const int   colc  = (col < NW) ? col : (NW - 1);   // clamped, loop-invariant
    const float nmask = (col < NW) ? 1.0f : 0.0f;

    v8f c = {};
    if (bias) {
        const float bv = bias[colc] * nmask;
        #pragma unroll
        for (int v = 0; v < 8; ++v) c[v] = bv;
    }

    const float* arow = A + (size_t)row * lda;
    const float* wcol = W + colc;
    for (int k = 0; k < Kpad; k += 4) {
        const int ka  = k + 2 * lg;
        const int k0  = (ka     < KA) ? ka     : (KA - 1);
        const int k1  = (ka + 1 < KA) ? ka + 1 : (KA - 1);
        const float m0 = (ka     < KA) ? 1.0f : 0.0f;
        const float m1 = (ka + 1 < KA) ? 1.0f : 0.0f;
        v2f a, b;
        a.x = arow[k0] * m0;
        a.y = arow[k1] * m1;
        b.x = wcol[(size_t)k0 * ldw] * (m0 * nmask);
        b.y = wcol[(size_t)k1 * ldw] * (m1 * nmask);
        c = wmma4(a, b, c);
    }

    const int crow0 = (tm << 4) + (lg << 3);
    if (act == 1) {                             // numerically-stable softplus
        #pragma unroll
        for (int v = 0; v < 8; ++v) {
            const float x = c[v];
            c[v] = fmaxf(x, 0.0f) + log1pf(__expf(-fabsf(x)));
        }
    }
    #pragma unroll
    for (int v = 0; v < 8; ++v)
        C[(size_t)(crow0 + v) * ldc + col] = c[v];
}

// ---------------------------------------------------------------------------
// Bilinear (half-pixel) upsample H x H -> 64 x 64, writing a 192-col slice of
// the 768-wide concat buffer. src: (BATCH, H, H, EMBED) row-major.
// ---------------------------------------------------------------------------
__global__ void resize_bilinear_into_cat(const float* __restrict__ src, int H,
                                         float* __restrict__ cat, int colOff)
{
    const int idx   = blockIdx.x * blockDim.x + threadIdx.x;
    const int total = BATCH * 64 * 64 * EMBED;
    if (idx >= total) return;
    const int e = idx % EMBED;
    int t = idx / EMBED;
    const int x = t % 64; t /= 64;
    const int y = t % 64;
    const int b = t / 64;

    const float scale = (float)H / 64.0f;
    const float sx = ((float)x + 0.5f) * scale - 0.5f;
    const float sy = ((float)y + 0.5f) * scale - 0.5f;
    int x0 = (int)floorf(sx); const float fx = sx - (float)x0;
    int y0 = (int)floorf(sy); const float fy = sy - (float)y0;
    int x1 = x0 + 1, y1 = y0 + 1;
    x0 = x0 < 0 ? 0 : (x0 > H - 1 ? H - 1 : x0);
    x1 = x1 < 0 ? 0 : (x1 > H - 1 ? H - 1 : x1);
    y0 = y0 < 0 ? 0 : (y0 > H - 1 ? H - 1 : y0);
    y1 = y1 < 0 ? 0 : (y1 > H - 1 ? H - 1 : y1);

    const float* sb = src + (size_t)b * H * H * EMBED;
    const float v00 = sb[((size_t)y0 * H + x0) * EMBED + e];
    const float v01 = sb[((size_t)y0 * H + x1) * EMBED + e];
    const float v10 = sb[((size_t)y1 * H + x0) * EMBED + e];
    const float v11 = sb[((size_t)y1 * H + x1) * EMBED + e];
    const float v = (1.0f - fy) * ((1.0f - fx) * v00 + fx * v01)
                  +          fy * ((1.0f - fx) * v10 + fx * v11);
    cat[((size_t)(b * 4096 + y * 64 + x)) * XZLD + colOff + e] = v;
}

// ---------------------------------------------------------------------------
// Row layernorm over EMBED=192 cols, one wave32 per row (6 cols/lane).
// ---------------------------------------------------------------------------
__global__ void layernorm_k(const float* __restrict__ x, const float* __restrict__ g,
                            const float* __restrict__ b, float* __restrict__ out,
                            int nrows)
{
    const int wave = (blockIdx.x * blockDim.x + threadIdx.x) >> 5;
    const int lane = threadIdx.x & 31;
    if (wave >= nrows) return;
    const float* xr = x + (size_t)wave * EMBED;
    float vals[6];
    float s = 0.0f, s2 = 0.0f;
    #pragma unroll
    for (int i = 0; i < 6; ++i) {
        const float v = xr[lane + 32 * i];
        vals[i] = v; s += v; s2 += v * v;
    }
    #pragma unroll
    for (int m = 16; m >= 1; m >>= 1) {
        s  += __shfl_xor(s,  m, 32);
        s2 += __shfl_xor(s2, m, 32);
    }
    const float mean = s  * (1.0f / EMBED);
    const float var  = s2 * (1.0f / EMBED) - mean * mean;
    const float r    = rsqrtf(var + 1e-5f);
    float* orow = out + (size_t)wave * EMBED;
    #pragma unroll
    for (int i = 0; i < 6; ++i) {
        const int cI = lane + 32 * i;
        orow[cI] = (vals[i] - mean) * r * g[cI] + b[cI];
    }
}

// ---------------------------------------------------------------------------
// Depthwise causal conv (width 4) over L on xi = xz[:, :, 0:384], + bias, SiLU.
// ---------------------------------------------------------------------------
__global__ void conv_silu_k(const float* __restrict__ xz, const float* __restrict__ cw,
                            const float* __restrict__ cb, float* __restrict__ u)
{
    const int idx   = blockIdx.x * blockDim.x + threadIdx.x;
    const int total = BATCH * LTOK * DIN;
    if (idx >= total) return;
    const int d  = idx % DIN;
    const int l  = (idx / DIN) % LTOK;
    const int bt = idx / (DIN * LTOK);
    float acc = cb[d];
    const float* base = xz + ((size_t)bt * LTOK) * XZLD + d;
    #pragma unroll
    for (int k = 0; k < DCONV; ++k) {
        const int ls = l - (DCONV - 1) + k;
        if (ls >= 0) acc += cw[d * DCONV + k] * base[(size_t)ls * XZLD];
    }
    u[idx] = acc / (1.0f + __expf(-acc));   // silu
}

// ---------------------------------------------------------------------------
// Selective scan. One block per (batch, 16-channel block); thread = (d, s).
// h_{l} = exp(dt*A[d,s]) * h_{l-1} + dt*u*B[l,s];  y[l,d] = sum_s h*C[l,s].
// dt/u/B/C chunk-staged through LDS; sum over s via 4 shfl_xor steps (<16).
// ---------------------------------------------------------------------------
#define SCAN_CH 64
__global__ void scan_k(const float* __restrict__ dt, const float* __restrict__ u,
                       const float* __restrict__ dbc, const float* __restrict__ Alog,
                       float* __restrict__ y)
{
    __shared__ float sdt[SCAN_CH][16];
    __shared__ float su [SCAN_CH][16];
    __shared__ float sB [SCAN_CH][16];
    __shared__ float sC [SCAN_CH][16];

    const int b    = blockIdx.x / (DIN / 16);
    const int dblk = blockIdx.x % (DIN / 16);
    const int d0   = dblk * 16;
    const int tid  = threadIdx.x;        // 256
    const int dl   = tid >> 4;           // 0..15 channel-in-block
    const int s    = tid & 15;           // state index
    const int d    = d0 + dl;

    const float Ads = -__expf(Alog[d * DSTATE + s]);
    float h = 0.0f;
    const size_t rowbase = (size_t)b * LTOK;

    for (int l0 = 0; l0 < LTOK; l0 += SCAN_CH) {
        #pragma unroll
        for (int it = tid; it < SCAN_CH * 16; it += 256) {
            const int j  = it >> 4;
            const int dd = it & 15;
            const size_t r = rowbase + l0 + j;
            sdt[j][dd] = dt [r * DIN   + d0 + dd];
            su [j][dd] = u  [r * DIN   + d0 + dd];
            sB [j][dd] = dbc[r * DBCLD + DTRANK + dd];
            sC [j][dd] = dbc[r * DBCLD + DTRANK + DSTATE + dd];
        }
        __syncthreads();
        for (int j = 0; j < SCAN_CH; ++j) {
            const float dtv = sdt[j][dl];
            const float uv  = su [j][dl];
            const float dA  = __expf(dtv * Ads);
            h = dA * h + (dtv * uv) * sB[j][s];
            float p = h * sC[j][s];
            p += __shfl_xor(p, 1, 32);
            p += __shfl_xor(p, 2, 32);
            p += __shfl_xor(p, 4, 32);
            p += __shfl_xor(p, 8, 32);
            if (s == 0) y[(rowbase + l0 + j) * DIN + d] = p;
        }
        __syncthreads();
    }
}

// ---------------------------------------------------------------------------
// y = (y + u * D_param[d]) * silu(z),  z = xz[:, :, 384+d]
// ---------------------------------------------------------------------------
__global__ void gate_k(float* __restrict__ y, const float* __restrict__ u,
                       const float* __restrict__ xz, const float* __restrict__ Dp)
{
    const int idx   = blockIdx.x * blockDim.x + threadIdx.x;
    const int total = BATCH * LTOK * DIN;
    if (idx >= total) return;
    const int d   = idx % DIN;
    const int row = idx / DIN;
    const float z  = xz[(size_t)row * XZLD + DIN + d];
    const float sz = z / (1.0f + __expf(-z));
    y[idx] = (y[idx] + u[idx] * Dp[d]) * sz;
}

// ---------------------------------------------------------------------------
extern "C" void kernel_launch(void* const* d_in, const int* in_sizes, int n_in,
                              void* d_out, int out_size, void* d_ws, size_t ws_size,
                              hipStream_t stream)
{
    (void)in_sizes; (void)n_in; (void)out_size; (void)ws_size;
    const float* tokens0    = (const float*)d_in[0];
    const float* tokens1    = (const float*)d_in[1];
    const float* tokens2    = (const float*)d_in[2];
    const float* tokens3    = (const float*)d_in[3];
    const float* proj_w0    = (const float*)d_in[4];
    const float* proj_b0    = (const float*)d_in[5];
    const float* proj_w1    = (const float*)d_in[6];
    const float* proj_b1    = (const float*)d_in[7];
    const float* proj_w2    = (const float*)d_in[8];
    const float* proj_b2    = (const float*)d_in[9];
    const float* proj_w3    = (const float*)d_in[10];
    const float* proj_b3    = (const float*)d_in[11];
    const float* fuse_w     = (const float*)d_in[12];
    const float* fuse_b     = (const float*)d_in[13];
    const float* ln_g       = (const float*)d_in[14];
    const float* ln_b       = (const float*)d_in[15];
    const float* in_proj_w  = (const float*)d_in[16];
    const float* conv_w     = (const float*)d_in[17];
    const float* conv_b     = (const float*)d_in[18];
    const float* x_proj_w   = (const float*)d_in[19];
    const float* dt_w       = (const float*)d_in[20];
    const float* dt_b       = (const float*)d_in[21];
    const float* A_log      = (const float*)d_in[22];
    const float* D_param    = (const float*)d_in[23];
    const float* out_proj_w = (const float*)d_in[24];

    const int ROWS = BATCH * LTOK;                 // 8192
    float* ws = (float*)d_ws;
    float* cat_xz = ws;                            // 8192*768  (concat, then xz)
    float* nat0   = cat_xz + (size_t)ROWS * XZLD;  // 2048*192  (later: dbc 8192*48)
    float* nat1   = nat0   + 2048 * EMBED;         // 512*192
    float* xln    = nat1   + 512 * EMBED;          // 8192*192
    float* ubuf   = xln    + (size_t)ROWS * EMBED; // 8192*384
    float* dtbuf  = ubuf   + (size_t)ROWS * DIN;   // 8192*384
    float* ybuf   = dtbuf  + (size_t)ROWS * DIN;   // 8192*384
    float* dbc    = nat0;                          // reuse after resize
    float* xbuf   = (float*)d_out;                 // residual stream x (2,4096,192)

    // fast path: 32x32 tile per wave, 8 waves per block
    auto gemm = [&](const float* A, int lda, const float* W, int ldw,
                    const float* bias, const float* res, int ldr,
                    float* C, int ldc, int M, int N, int K) {
        const int tiles  = (M / 32) * (N / 32);
        const int blocks = (tiles + 7) / 8;
        wmma_gemm_f32_4t<<<blocks, 256, 0, stream>>>(A, lda, W, ldw, bias, res,
                                                     ldr, C, ldc, M, N, K);
    };
    // guarded path: 16x16 tile per wave
    auto gemmg = [&](const float* A, int lda, const float* W, int ldw,
                     const float* bias, float* C, int ldc,
                     int M, int N, int Kpad, int KA, int NW, int act) {
        const int tiles  = (M / 16) * (N / 16);
        const int blocks = (tiles + 7) / 8;
        wmma_gemm_f32_guard<<<blocks, 256, 0, stream>>>(A, lda, W, ldw, bias,
                                                        C, ldc, M, N, Kpad, KA,
                                                        NW, act);
    };

    // ---- stage 1: per-stream projections -------------------------------
    gemm(tokens0, 192, proj_w0, EMBED, proj_b0, nullptr, 0, nat0, EMBED,
         BATCH * 1024, EMBED, 192);
    gemm(tokens1, 192, proj_w1, EMBED, proj_b1, nullptr, 0, nat1, EMBED,
         BATCH * 256, EMBED, 192);
    gemm(tokens2,  96, proj_w2, EMBED, proj_b2, nullptr, 0, cat_xz + 2 * EMBED, XZLD,
         ROWS, EMBED, 96);
    gemm(tokens3,  96, proj_w3, EMBED, proj_b3, nullptr, 0, cat_xz + 3 * EMBED, XZLD,
         ROWS, EMBED, 96);

    // ---- stage 2: bilinear align of streams 0/1 into concat ------------
    const int rtot = BATCH * 64 * 64 * EMBED;
    resize_bilinear_into_cat<<<(rtot + 255) / 256, 256, 0, stream>>>(nat0, 32, cat_xz, 0);
    resize_bilinear_into_cat<<<(rtot + 255) / 256, 256, 0, stream>>>(nat1, 16, cat_xz, EMBED);

    // ---- stage 3: fuse GEMM (768 -> 192) into residual stream x --------
    gemm(cat_xz, XZLD, fuse_w, EMBED, fuse_b, nullptr, 0, xbuf, EMBED,
         ROWS, EMBED, 768);

    // ---- stage 4: 3 Mamba blocks ---------------------------------------
    for (int i = 0; i < 3; ++i) {
        const float* lg  = ln_g + i * EMBED;
        const float* lb  = ln_b + i * EMBED;
        const float* ipw = in_proj_w  + (size_t)i * EMBED * XZLD;
        const float* cw  = conv_w     + (size_t)i * DIN * DCONV;
        const float* cb  = conv_b     + (size_t)i * DIN;
        const float* xpw = x_proj_w   + (size_t)i * DIN * (DTRANK + 2 * DSTATE);
        const float* dtw = dt_w       + (size_t)i * DTRANK * DIN;
        const float* dtbias = dt_b    + (size_t)i * DIN;
        const float* al  = A_log      + (size_t)i * DIN * DSTATE;
        const float* dp  = D_param    + (size_t)i * DIN;
        const float* opw = out_proj_w + (size_t)i * DIN * EMBED;

        layernorm_k<<<ROWS / 8, 256, 0, stream>>>(xbuf, lg, lb, xln, ROWS);

        // xz = xln @ in_proj_w  (192 -> 768); xi = cols 0..383, z = 384..767
        gemm(xln, EMBED, ipw, XZLD, nullptr, nullptr, 0, cat_xz, XZLD,
             ROWS, XZLD, 192);

        const int etot = ROWS * DIN;
        conv_silu_k<<<(etot + 255) / 256, 256, 0, stream>>>(cat_xz, cw, cb, ubuf);

        // dbc = u @ x_proj_w  (384 -> 44, padded to 48 cols)
        gemmg(ubuf, DIN, xpw, DTRANK + 2 * DSTATE, nullptr, dbc, DBCLD,
              ROWS, DBCLD, 384, 384, DTRANK + 2 * DSTATE, 0);

        // dt = softplus(dbc[:, :12] @ dt_w + dt_b)  (K padded 12 -> 16)
        gemmg(dbc, DBCLD, dtw, DIN, dtbias, dtbuf, DIN,
              ROWS, DIN, 16, DTRANK, DIN, 1);

        scan_k<<<BATCH * (DIN / 16), 256, 0, stream>>>(dtbuf, ubuf, dbc, al, ybuf);

        gate_k<<<(etot + 255) / 256, 256, 0, stream>>>(ybuf, ubuf, cat_xz, dp);

        // x = x + y @ out_proj_w  (384 -> 192), residual add fused in GEMM
        gemm(ybuf, DIN, opw, EMBED, nullptr, xbuf, EMBED, xbuf, EMBED,
             ROWS, EMBED, 384);
    }
}